// Stage2GNN_57483842290347
// MI455X (gfx1250) — compile-verified
//
#include <hip/hip_runtime.h>

// ---------------------------------------------------------------------------
// Stage2 GNN (2x SAGEConv + 3 heads) for MI455X / gfx1250, wave32.
// Dense per-node projections use V_WMMA_F32_16X16X4_F32 accumulation chains
// (full fp32 precision, matches reference). Edge aggregation uses
// global_atomic_add_f32 scatter (memory-bound phase).
// Full 16-node tiles take a branch-free path (GUARD=false); at most one
// remainder tile runs the guarded instantiation.
// ---------------------------------------------------------------------------

typedef float v2f __attribute__((ext_vector_type(2)));
typedef float v8f __attribute__((ext_vector_type(8)));

#define WMMA_F32(a, b, c) \
  __builtin_amdgcn_wmma_f32_16x16x4_f32(false, (a), false, (b), (short)0, (c), false, false)

#define IN_DIM 15
#define HID    32

// ---------------------------------------------------------------------------
__global__ __launch_bounds__(256) void gnn_zero_kernel(float* __restrict__ p, size_t n) {
  size_t i = (size_t)blockIdx.x * blockDim.x + threadIdx.x;
  size_t stride = (size_t)gridDim.x * blockDim.x;
  for (; i < n; i += stride) p[i] = 0.0f;
}

// Layer-1 edge scatter: 16 threads per edge; f==15 thread counts degree.
__global__ __launch_bounds__(256) void gnn_scatter1_kernel(
    const int* __restrict__ src, const int* __restrict__ dst,
    const float* __restrict__ x, float* __restrict__ agg,
    float* __restrict__ deg, int E) {
  long long idx = (long long)blockIdx.x * blockDim.x + threadIdx.x;
  if (idx >= (long long)E * 16) return;
  int e = (int)(idx >> 4);
  int f = (int)(idx & 15);
  int d = dst[e];
  if (f == 15) {
    atomicAdd(deg + d, 1.0f);
  } else {
    int s = src[e];
    atomicAdd(agg + (size_t)d * 16 + f, x[(size_t)s * IN_DIM + f]);
  }
}

// Layer-2 edge scatter: 32 threads per edge (coalesced row gather + scatter).
__global__ __launch_bounds__(256) void gnn_scatter2_kernel(
    const int* __restrict__ src, const int* __restrict__ dst,
    const float* __restrict__ h, float* __restrict__ agg, int E) {
  long long idx = (long long)blockIdx.x * blockDim.x + threadIdx.x;
  if (idx >= (long long)E * 32) return;
  int e = (int)(idx >> 5);
  int f = (int)(idx & 31);
  int s = src[e];
  int d = dst[e];
  atomicAdd(agg + (size_t)d * HID + f, h[(size_t)s * HID + f]);
}

// ---------------------------------------------------------------------------
// SAGE layer 1: h1 = relu([mean1 | x] @ [w1_l ; w1_r]^T + b1_l)
// One wave per 16-node tile; K padded 30 -> 32 (8 chained wmma per col tile).
template <bool GUARD>
__global__ __launch_bounds__(256) void gnn_sage1_kernel(
    const float* __restrict__ x, const float* __restrict__ agg,
    const float* __restrict__ deg, const float* __restrict__ wl,
    const float* __restrict__ bl, const float* __restrict__ wr,
    float* __restrict__ hout, int tile0, int tile_end, int N) {
  const int lane = threadIdx.x & 31;
  const int wave = threadIdx.x >> 5;
  const int tile = tile0 + blockIdx.x * 8 + wave;
  if (tile >= tile_end) return;  // wave-uniform
  const int row0 = tile * 16;
  const int m    = lane & 15;
  const int half = lane >> 4;
  const int node = row0 + m;
  const bool valid = GUARD ? (node < N) : true;
  const int nc = GUARD ? (valid ? node : N - 1) : node;
  const float inv = 1.0f / fmaxf(deg[nc], 1.0f);

  auto aval = [&](int k) -> float {  // A[m][k], K = [mean(16) | x(16)]
    float v;
    if (k < 16) {
      int kc = (k < IN_DIM) ? k : 0;
      v = agg[(size_t)nc * 16 + kc] * inv;
      v = (k < IN_DIM) ? v : 0.0f;
    } else {
      int kc = k - 16;
      int kcc = (kc < IN_DIM) ? kc : 0;
      v = x[(size_t)nc * IN_DIM + kcc];
      v = (kc < IN_DIM) ? v : 0.0f;
    }
    return valid ? v : 0.0f;
  };
  auto bval = [&](int k, int n) -> float {  // B[k][n] = stacked [w_l ; w_r]^T
    if (k < 16) {
      int kc = (k < IN_DIM) ? k : 0;
      float v = wl[n * IN_DIM + kc];
      return (k < IN_DIM) ? v : 0.0f;
    } else {
      int kc = k - 16;
      int kcc = (kc < IN_DIM) ? kc : 0;
      float v = wr[n * IN_DIM + kcc];
      return (kc < IN_DIM) ? v : 0.0f;
    }
  };

#pragma unroll
  for (int nb = 0; nb < 2; ++nb) {
    const int n = nb * 16 + m;
    v8f c = {0.0f, 0.0f, 0.0f, 0.0f, 0.0f, 0.0f, 0.0f, 0.0f};
#pragma unroll
    for (int kk = 0; kk < 8; ++kk) {
      const int k0 = kk * 4 + 2 * half;
      v2f a, b;
      a.x = aval(k0);      a.y = aval(k0 + 1);
      b.x = bval(k0, n);   b.y = bval(k0 + 1, n);
      c = WMMA_F32(a, b, c);
    }
    const float bias = bl[n];
#pragma unroll
    for (int v = 0; v < 8; ++v) {
      const int r = row0 + v + 8 * half;  // D row = vgpr + 8*half, col = m
      const float val = fmaxf(c[v] + bias, 0.0f);
      if (!GUARD || r < N) hout[(size_t)r * HID + n] = val;
    }
  }
}

// SAGE layer 2: h2 = relu([mean2 | h1] @ [w2_l ; w2_r]^T + b2_l), K = 64.
template <bool GUARD>
__global__ __launch_bounds__(256) void gnn_sage2_kernel(
    const float* __restrict__ h1, const float* __restrict__ agg,
    const float* __restrict__ deg, const float* __restrict__ wl,
    const float* __restrict__ bl, const float* __restrict__ wr,
    float* __restrict__ hout, int tile0, int tile_end, int N) {
  const int lane = threadIdx.x & 31;
  const int wave = threadIdx.x >> 5;
  const int tile = tile0 + blockIdx.x * 8 + wave;
  if (tile >= tile_end) return;
  const int row0 = tile * 16;
  const int m    = lane & 15;
  const int half = lane >> 4;
  const int node = row0 + m;
  const bool valid = GUARD ? (node < N) : true;
  const int nc = GUARD ? (valid ? node : N - 1) : node;
  const float inv = 1.0f / fmaxf(deg[nc], 1.0f);

  auto aval = [&](int k) -> float {
    float v = (k < HID) ? agg[(size_t)nc * HID + k] * inv
                        : h1[(size_t)nc * HID + (k - HID)];
    return valid ? v : 0.0f;
  };
  auto bval = [&](int k, int n) -> float {
    return (k < HID) ? wl[n * HID + k] : wr[n * HID + (k - HID)];
  };

#pragma unroll
  for (int nb = 0; nb < 2; ++nb) {
    const int n = nb * 16 + m;
    v8f c = {0.0f, 0.0f, 0.0f, 0.0f, 0.0f, 0.0f, 0.0f, 0.0f};
#pragma unroll
    for (int kk = 0; kk < 16; ++kk) {
      const int k0 = kk * 4 + 2 * half;
      v2f a, b;
      a.x = aval(k0);      a.y = aval(k0 + 1);
      b.x = bval(k0, n);   b.y = bval(k0 + 1, n);
      c = WMMA_F32(a, b, c);
    }
    const float bias = bl[n];
#pragma unroll
    for (int v = 0; v < 8; ++v) {
      const int r = row0 + v + 8 * half;
      const float val = fmaxf(c[v] + bias, 0.0f);
      if (!GUARD || r < N) hout[(size_t)r * HID + n] = val;
    }
  }
}

// Heads: one 16-col D tile covers [head(3) | mat(8) | len(1) | pad(4)].
template <bool GUARD>
__global__ __launch_bounds__(256) void gnn_heads_kernel(
    const float* __restrict__ h2, const float* __restrict__ wh,
    const float* __restrict__ bh, const float* __restrict__ wm,
    const float* __restrict__ bm, const float* __restrict__ wlen,
    const float* __restrict__ blen, float* __restrict__ out,
    int tile0, int tile_end, int N) {
  const int lane = threadIdx.x & 31;
  const int wave = threadIdx.x >> 5;
  const int tile = tile0 + blockIdx.x * 8 + wave;
  if (tile >= tile_end) return;
  const int row0 = tile * 16;
  const int m    = lane & 15;
  const int half = lane >> 4;
  const int node = row0 + m;
  const bool valid = GUARD ? (node < N) : true;
  const int nc = GUARD ? (valid ? node : N - 1) : node;

  auto bval = [&](int k, int n) -> float {
    if (n < 3)  return wh[n * HID + k];
    if (n < 11) return wm[(n - 3) * HID + k];
    if (n == 11) return wlen[k];
    return 0.0f;
  };

  v8f c = {0.0f, 0.0f, 0.0f, 0.0f, 0.0f, 0.0f, 0.0f, 0.0f};
#pragma unroll
  for (int kk = 0; kk < 8; ++kk) {
    const int k0 = kk * 4 + 2 * half;
    v2f a, b;
    float a0 = h2[(size_t)nc * HID + k0];
    float a1 = h2[(size_t)nc * HID + k0 + 1];
    a.x = valid ? a0 : 0.0f;
    a.y = valid ? a1 : 0.0f;
    b.x = bval(k0, m);
    b.y = bval(k0 + 1, m);
    c = WMMA_F32(a, b, c);
  }

  const size_t off_mat = (size_t)3 * N;
  const size_t off_len = (size_t)11 * N;
#pragma unroll
  for (int v = 0; v < 8; ++v) {
    const int r = row0 + v + 8 * half;
    if (!GUARD || r < N) {
      const float val = c[v];
      if (m < 3)        out[(size_t)r * 3 + m] = val + bh[m];
      else if (m < 11)  out[off_mat + (size_t)r * 8 + (m - 3)] = val + bm[m - 3];
      else if (m == 11) out[off_len + r] = val + blen[0];
    }
  }
}

// ---------------------------------------------------------------------------
extern "C" void kernel_launch(void* const* d_in, const int* in_sizes, int n_in,
                              void* d_out, int out_size, void* d_ws, size_t ws_size,
                              hipStream_t stream) {
  const float* x   = (const float*)d_in[0];
  const int*   ei  = (const int*)d_in[1];
  const float* w1l = (const float*)d_in[2];
  const float* b1l = (const float*)d_in[3];
  const float* w1r = (const float*)d_in[4];
  const float* w2l = (const float*)d_in[5];
  const float* b2l = (const float*)d_in[6];
  const float* w2r = (const float*)d_in[7];
  const float* wh  = (const float*)d_in[8];
  const float* bh  = (const float*)d_in[9];
  const float* wm  = (const float*)d_in[10];
  const float* bm  = (const float*)d_in[11];
  const float* wln = (const float*)d_in[12];
  const float* bln = (const float*)d_in[13];
  float* out = (float*)d_out;

  const int N = in_sizes[0] / IN_DIM;
  const int E = in_sizes[1] / 2;
  const int* src = ei;
  const int* dst = ei + E;

  // Workspace layout (floats): deg[N] | agg1[16N] | agg2[32N] | h1[32N] | h2[32N]
  float* ws   = (float*)d_ws;
  float* deg  = ws;
  float* agg1 = ws + (size_t)N;
  float* agg2 = ws + (size_t)17 * N;
  float* h1   = ws + (size_t)49 * N;
  float* h2   = ws + (size_t)81 * N;

  // 0) zero deg + agg1 + agg2 (contiguous 49N floats)
  gnn_zero_kernel<<<2048, 256, 0, stream>>>(ws, (size_t)49 * N);

  // 1) layer-1 edge scatter (16 threads/edge)
  {
    long long total = (long long)E * 16;
    int blocks = (int)((total + 255) / 256);
    gnn_scatter1_kernel<<<blocks, 256, 0, stream>>>(src, dst, x, agg1, deg, E);
  }

  const int tiles_total = (N + 15) / 16;
  const int tiles_full  = N / 16;           // branch-free tiles
  const bool has_tail   = tiles_full != tiles_total;
  const int full_blocks = (tiles_full + 7) / 8;

  // 2) layer-1 node update (WMMA f32)
  if (tiles_full > 0)
    gnn_sage1_kernel<false><<<full_blocks, 256, 0, stream>>>(
        x, agg1, deg, w1l, b1l, w1r, h1, 0, tiles_full, N);
  if (has_tail)
    gnn_sage1_kernel<true><<<1, 256, 0, stream>>>(
        x, agg1, deg, w1l, b1l, w1r, h1, tiles_full, tiles_total, N);

  // 3) layer-2 edge scatter (32 threads/edge)
  {
    long long total = (long long)E * 32;
    int blocks = (int)((total + 255) / 256);
    gnn_scatter2_kernel<<<blocks, 256, 0, stream>>>(src, dst, h1, agg2, E);
  }

  // 4) layer-2 node update (WMMA f32)
  if (tiles_full > 0)
    gnn_sage2_kernel<false><<<full_blocks, 256, 0, stream>>>(
        h1, agg2, deg, w2l, b2l, w2r, h2, 0, tiles_full, N);
  if (has_tail)
    gnn_sage2_kernel<true><<<1, 256, 0, stream>>>(
        h1, agg2, deg, w2l, b2l, w2r, h2, tiles_full, tiles_total, N);

  // 5) output heads (WMMA f32, single col-tile scattered to 3 outputs)
  if (tiles_full > 0)
    gnn_heads_kernel<false><<<full_blocks, 256, 0, stream>>>(
        h2, wh, bh, wm, bm, wln, bln, out, 0, tiles_full, N);
  if (has_tail)
    gnn_heads_kernel<true><<<1, 256, 0, stream>>>(
        h2, wh, bh, wm, bm, wln, bln, out, tiles_full, tiles_total, N);
}